// SparsePooling_1632087572807
// MI455X (gfx1250) — compile-verified
//
#include <hip/hip_runtime.h>

typedef __attribute__((ext_vector_type(2))) float v2f;
typedef __attribute__((ext_vector_type(8))) float v8f;

#define NUM_GRAPHS 512
#define NPG 256          // nodes per graph
#define KPG 128          // selected per graph
#define DFEAT 128
#define DEDGE 16
#define NUM_EDGES 4194304
#define NNODES (NUM_GRAPHS * NPG)   // 131072
#define NSEL   (NUM_GRAPHS * KPG)   // 65536

// ---- output layout (flat, reference return order) ----
#define OUT_XOUT  0
#define OUT_EI    (NSEL * DFEAT)                   // 8388608
#define OUT_ATTR  (OUT_EI + 2 * NUM_EDGES)         // 16777216
#define OUT_BATCH (OUT_ATTR + NUM_EDGES * DEDGE)   // 83886080
#define OUT_PERM  (OUT_BATCH + NSEL)               // 83951616
#define OUT_W     (OUT_PERM + NSEL)                // 84017152

// ---- workspace layout (int32 slots) ----
#define WS_NODEMAP 0
#define WS_BLKCNT  NNODES
#define WS_BLKBASE (WS_BLKCNT + 1024)
#define WS_TOTAL   (WS_BLKBASE + 1024)

#define EPB 4096                 // edges per block
#define NBLK (NUM_EDGES / EPB)   // 1024
#define EPT 16                   // edges per thread (256 threads/block)

// ------------------------------------------------------------------
// Kernel A: per-graph scoring (f32 WMMA), stable top-k, gather+scale.
// One 256-thread block (8 wave32) per graph.
// ------------------------------------------------------------------
__global__ __launch_bounds__(256)
void topk_score_gather(const float* __restrict__ xg,
                       const float* __restrict__ Wv,
                       const float* __restrict__ bv,
                       float* __restrict__ out,
                       int* __restrict__ node_map) {
    __shared__ float Ws[DFEAT];
    __shared__ float sc[NPG];
    __shared__ int   selRow[KPG];
    __shared__ float selW[KPG];

    const int g     = blockIdx.x;
    const int tid   = threadIdx.x;
    const int lane  = tid & 31;
    const int wave  = tid >> 5;
    const int gbase = g * NPG;

    if (tid < DFEAT) Ws[tid] = Wv[tid];
    __syncthreads();
    const float bval = bv[0];

    // ---- scores via V_WMMA_F32_16X16X4_F32 ----
    // A (16x4 f32): lanes 0-15 hold K=0,1; lanes 16-31 hold K=2,3.
    // B (4x16 f32): broadcast W[k0+kk] across all 16 columns, so every
    // column of D holds the dot product; extract column 0 (lanes 0 / 16).
    const int koff = (lane < 16) ? 0 : 2;
    for (int t = 0; t < 2; ++t) {
        const int m0   = (wave * 2 + t) * 16;
        const int mrow = gbase + m0 + (lane & 15);
        const float2* rp = (const float2*)(xg + (size_t)mrow * DFEAT);
        v8f acc = {};
#pragma unroll
        for (int k0 = 0; k0 < DFEAT; k0 += 4) {
            float2 av = rp[(k0 + koff) >> 1];
            v2f A = {av.x, av.y};
            v2f B = {Ws[k0 + koff], Ws[k0 + koff + 1]};
            acc = __builtin_amdgcn_wmma_f32_16x16x4_f32(
                false, A, false, B, (short)0, acc, false, false);
        }
        // D layout: VGPR r -> (M=r, N=lane) for lanes 0-15; (M=8+r, N=lane-16)
        // for lanes 16-31. Column N=0 lives in lanes 0 and 16.
        if ((lane & 15) == 0) {
            const int mb = m0 + ((lane >> 4) << 3);
#pragma unroll
            for (int r = 0; r < 8; ++r) sc[mb + r] = acc[r] + bval;
        }
    }
    __syncthreads();

    // ---- stable descending rank (matches lax.top_k tie-breaking) ----
    {
        const int   i  = tid;
        const float si = sc[i];
        int rank = 0;
        for (int j = 0; j < NPG; ++j) {
            const float sj = sc[j];
            rank += (sj > si) || (sj == si && j < i);
        }
        const int gid = gbase + i;
        if (rank < KPG) {
            const int p = g * KPG + rank;
            node_map[gid] = p;
            selRow[rank]  = i;
            selW[rank]    = si;
            ((int*)out)[OUT_PERM + p]  = gid;
            out[OUT_W + p]             = si;
            ((int*)out)[OUT_BATCH + p] = g;   // batch[perm] == graph id
        } else {
            node_map[gid] = -1;
        }
    }
    __syncthreads();

    // ---- x_out = x[perm] * w (float4, coalesced; x is L2-resident) ----
    const float4* xg4  = (const float4*)xg;
    float4*       out4 = (float4*)(out + OUT_XOUT);
    const int QPR = DFEAT / 4;   // 32 float4 per row
    for (int idx = tid; idx < KPG * QPR; idx += 256) {
        const int p = idx / QPR;
        const int q = idx - p * QPR;
        const int row = gbase + selRow[p];
        const float wv = selW[p];
        float4 v = xg4[(size_t)row * QPR + q];
        v.x *= wv; v.y *= wv; v.z *= wv; v.w *= wv;
        out4[(size_t)(g * KPG + p) * QPR + q] = v;
    }
}

// ------------------------------------------------------------------
// Kernel E1: per-block kept-edge counts (coalesced strided access).
// ------------------------------------------------------------------
__global__ __launch_bounds__(256)
void edge_count(const int* __restrict__ src, const int* __restrict__ dst,
                const int* __restrict__ node_map, int* __restrict__ blkcnt) {
    __shared__ int red[256];
    const int tid = threadIdx.x;
    const int e0  = blockIdx.x * EPB;
    int c = 0;
#pragma unroll
    for (int i = 0; i < EPT; ++i) {
        const int e = e0 + i * 256 + tid;
        c += (node_map[src[e]] >= 0) && (node_map[dst[e]] >= 0);
    }
    red[tid] = c;
    __syncthreads();
    for (int off = 128; off > 0; off >>= 1) {
        if (tid < off) red[tid] += red[tid + off];
        __syncthreads();
    }
    if (tid == 0) blkcnt[blockIdx.x] = red[0];
}

// ------------------------------------------------------------------
// Kernel E2: single-block exclusive scan of 1024 block counts.
// ------------------------------------------------------------------
__global__ __launch_bounds__(1024)
void edge_scan(const int* __restrict__ blkcnt,
               int* __restrict__ blkbase, int* __restrict__ total) {
    __shared__ int s[NBLK];
    const int t = threadIdx.x;
    const int v = blkcnt[t];
    s[t] = v;
    __syncthreads();
    for (int off = 1; off < NBLK; off <<= 1) {
        const int add = (t >= off) ? s[t - off] : 0;
        __syncthreads();
        s[t] += add;
        __syncthreads();
    }
    blkbase[t] = s[t] - v;                 // exclusive base
    if (t == NBLK - 1) *total = s[t];      // total kept
}

// ------------------------------------------------------------------
// Kernel E3: stable-compaction emit. Kept edges -> front in original
// order (remapped ids + attrs); dropped edges -> tail (-1 ids, 0 attrs).
// Each thread owns 16 consecutive edges so local ranks stay stable.
// ------------------------------------------------------------------
__global__ __launch_bounds__(256)
void edge_emit(const int* __restrict__ src, const int* __restrict__ dst,
               const float4* __restrict__ eattr4,
               const int* __restrict__ node_map,
               const int* __restrict__ blkbase,
               const int* __restrict__ totalp,
               float* __restrict__ out) {
    __shared__ int tc[256];
    const int tid = threadIdx.x;
    const int blk = blockIdx.x;
    const int e0  = blk * EPB + tid * EPT;

    int ns[EPT], nd[EPT];
    unsigned maskbits = 0;
#pragma unroll
    for (int i = 0; i < EPT; ++i) {
        const int e = e0 + i;
        const int a = node_map[src[e]];
        const int b = node_map[dst[e]];
        const bool k = (a >= 0) && (b >= 0);
        ns[i] = a; nd[i] = b;
        maskbits |= (unsigned)k << i;
        if (k) __builtin_prefetch((const void*)(eattr4 + (size_t)e * 4), 0, 0);
    }
    const int c = __popc(maskbits);
    tc[tid] = c;
    __syncthreads();
    // inclusive scan over 256 per-thread counts
    for (int off = 1; off < 256; off <<= 1) {
        const int add = (tid >= off) ? tc[tid - off] : 0;
        __syncthreads();
        tc[tid] += add;
        __syncthreads();
    }
    const int keptBaseBlk = blkbase[blk];
    const int total       = *totalp;
    int keptPos = keptBaseBlk + tc[tid] - c;           // my first kept slot
    int dropPos = total + (e0 - keptPos);              // droppedBefore = e0 - keptBefore

    int*    ei0   = (int*)out + OUT_EI;
    int*    ei1   = ei0 + NUM_EDGES;
    float4* attr4 = (float4*)(out + OUT_ATTR);

#pragma unroll
    for (int i = 0; i < EPT; ++i) {
        const int e = e0 + i;
        if ((maskbits >> i) & 1u) {
            const int p = keptPos++;
            ei0[p] = ns[i];
            ei1[p] = nd[i];
            const float4* a = eattr4 + (size_t)e * 4;
            float4* o = attr4 + (size_t)p * 4;
            o[0] = a[0]; o[1] = a[1]; o[2] = a[2]; o[3] = a[3];
        } else {
            const int p = dropPos++;
            ei0[p] = -1;
            ei1[p] = -1;
            const float4 z = {0.f, 0.f, 0.f, 0.f};
            float4* o = attr4 + (size_t)p * 4;
            o[0] = z; o[1] = z; o[2] = z; o[3] = z;
        }
    }
}

// ------------------------------------------------------------------
extern "C" void kernel_launch(void* const* d_in, const int* in_sizes, int n_in,
                              void* d_out, int out_size, void* d_ws, size_t ws_size,
                              hipStream_t stream) {
    const float* x     = (const float*)d_in[0];
    const int*   ei    = (const int*)  d_in[1];
    const float* eattr = (const float*)d_in[2];
    // d_in[3] = batch (unused: new_batch == graph id by construction)
    const float* W     = (const float*)d_in[4];
    const float* b     = (const float*)d_in[5];

    float* out = (float*)d_out;
    int*   ws  = (int*)d_ws;
    int* node_map = ws + WS_NODEMAP;
    int* blkcnt   = ws + WS_BLKCNT;
    int* blkbase  = ws + WS_BLKBASE;
    int* total    = ws + WS_TOTAL;

    const int* src = ei;
    const int* dst = ei + NUM_EDGES;

    topk_score_gather<<<dim3(NUM_GRAPHS), dim3(256), 0, stream>>>(
        x, W, b, out, node_map);
    edge_count<<<dim3(NBLK), dim3(256), 0, stream>>>(src, dst, node_map, blkcnt);
    edge_scan<<<dim3(1), dim3(NBLK), 0, stream>>>(blkcnt, blkbase, total);
    edge_emit<<<dim3(NBLK), dim3(256), 0, stream>>>(
        src, dst, (const float4*)eattr, node_map, blkbase, total, out);
}